// PhyGeoGrapH_10084583211165
// MI455X (gfx1250) — compile-verified
//
#include <hip/hip_runtime.h>
#include <cstddef>
#include <cstdint>

typedef __attribute__((ext_vector_type(2))) float v2f;
typedef __attribute__((ext_vector_type(8))) float v8f;

#define EPSBN 1e-5f

// ---------------- workspace layout (float offsets) ----------------
static const size_t F_R0  = 0;          // 12.8M: agg1/2/3, later zA [25000x512]
static const size_t F_R1  = 12800000;   // 6.4M : h1 [100000x64], later zB [25000x256]
static const size_t F_R2  = 19200000;   // 3.2M : h2 [50000x64],  later zC [25000x128]
static const size_t F_ZD  = 22400000;   // 6.4M : zD [25000x256]
static const size_t F_ZE  = 28800000;   // 12.8M: zE [25000x512]
static const size_t F_XIN = 41600000;   // 1.675M: xin [25000x67]
static const size_t F_TMP = 43275008;   // 1.675M: logits/prob/y [25000x67]
static const size_t F_ZF  = 44950016;   // 1.675M: zF [25000x67]
static const size_t F_H3  = 46625024;   // 25000
static const size_t F_XL1 = 46650024;   // 25000
static const size_t F_XL2 = 46675024;   // 25000
static const size_t F_CNT = 46700032;   // 100000
static const size_t F_SUM = 46800032;   // 512
static const size_t F_SSQ = 46800544;   // 512

// ---------------- CDNA5 async global->LDS copy (ASYNCcnt path) ----------------
// the LDS address is taken from the VDST VGPR; generic LDS pointers carry the LDS byte
// offset in their low 32 bits, so truncation gives the VDST lane value.
__device__ __forceinline__ void async_load_f32_to_lds(const float* gptr, float* lptr) {
  unsigned off = (unsigned)(unsigned long long)(uintptr_t)lptr;
  asm volatile("global_load_async_to_lds_b32 %0, %1, off"
               :
               : "v"(off), "v"(gptr)
               : "memory");
}
__device__ __forceinline__ void wait_async_all() {
  asm volatile("s_wait_asynccnt 0x0" ::: "memory");
}

// ---------------- utility kernels ----------------
__global__ __launch_bounds__(256) void k_zero(float* __restrict__ p, int n) {
  int i = blockIdx.x * 256 + threadIdx.x;
  if (i < n) p[i] = 0.f;
}

// Edge scatter: per (edge, feature) atomic accumulate of neighbor features,
// plus per-edge degree count. Feature width fixed at 64.
__global__ __launch_bounds__(256) void k_scatter(const int* __restrict__ src,
                                                 const int* __restrict__ dst,
                                                 const float* __restrict__ h,
                                                 float* __restrict__ agg,
                                                 float* __restrict__ cnt, int E) {
  int idx = blockIdx.x * 256 + threadIdx.x;
  int e = idx >> 6, f = idx & 63;
  if (e < E) {
    int s = src[e], t = dst[e];
    atomicAdd(&agg[(size_t)t * 64 + f], h[(size_t)s * 64 + f]);
    if (f == 0) atomicAdd(&cnt[t], 1.f);
  }
}

// ---------------- fused conv GEMM (WMMA fp32, N=K=64) ----------------
// out = act( (agg/max(cnt,1)) @ Wn + hroot @ Wr + b ).
// Block = 8 M-tiles x 1 N-tile; Wn/Wr panels staged in LDS via async copies.
__global__ __launch_bounds__(256) void k_conv_wmma(
    const float* __restrict__ agg, const float* __restrict__ cnt,
    const float* __restrict__ hroot, const float* __restrict__ Wn,
    const float* __restrict__ Wr, const float* __restrict__ bias,
    float* __restrict__ out, int M, int relu) {
  __shared__ float sWn[64 * 16];
  __shared__ float sWr[64 * 16];
  const int tilesN = 4;                                   // N = 64
  int n0 = (blockIdx.x % tilesN) * 16;
  int bm = blockIdx.x / tilesN;
  for (int idx = threadIdx.x; idx < 64 * 16; idx += 256) {
    int k = idx >> 4, c = idx & 15;
    async_load_f32_to_lds(&Wn[(size_t)k * 64 + n0 + c], &sWn[idx]);
    async_load_f32_to_lds(&Wr[(size_t)k * 64 + n0 + c], &sWr[idx]);
  }
  wait_async_all();
  __syncthreads();

  int tm = bm * 8 + (threadIdx.x >> 5);
  if (tm * 16 >= M) return;
  int lane = threadIdx.x & 31;
  int half = lane >> 4, l16 = lane & 15;
  int mrow = tm * 16 + l16;
  int ncol = n0 + l16;
  bool mok = mrow < M;
  float rinv = mok ? 1.f / fmaxf(cnt[mrow], 1.f) : 0.f;
  const float* aRow = agg + (size_t)mrow * 64;
  const float* rRow = hroot + (size_t)mrow * 64;
  v8f acc = {0.f, 0.f, 0.f, 0.f, 0.f, 0.f, 0.f, 0.f};
#pragma unroll 4
  for (int kb = 0; kb < 64; kb += 4) {
    int kk = kb + 2 * half;               // ISA A/B layout: lanes16-31 => K+2
    v2f a = {0.f, 0.f}, r = {0.f, 0.f};
    if (mok) {
      a = *(const v2f*)(aRow + kk);       // rows 64-wide -> 8B aligned pairs
      r = *(const v2f*)(rRow + kk);
      a = a * rinv;
    }
    v2f bn, br;
    bn[0] = sWn[kk * 16 + l16]; bn[1] = sWn[(kk + 1) * 16 + l16];
    br[0] = sWr[kk * 16 + l16]; br[1] = sWr[(kk + 1) * 16 + l16];
    acc = __builtin_amdgcn_wmma_f32_16x16x4_f32(false, a, false, bn, (short)0, acc, false, false);
    acc = __builtin_amdgcn_wmma_f32_16x16x4_f32(false, r, false, br, (short)0, acc, false, false);
  }
  float bb = bias[ncol];
#pragma unroll
  for (int rr = 0; rr < 8; ++rr) {
    int m = tm * 16 + rr + 8 * half;      // C/D layout: VGPR rr, lanes16-31 => M+8
    if (m < M) {
      float v = acc[rr] + bb;
      if (relu) v = fmaxf(v, 0.f);
      out[(size_t)m * 64 + ncol] = v;
    }
  }
}

// ---------------- generic WMMA fp32 GEMM: C = act(A @ W + b) ----------------
// Block = 8 M-tiles x 1 N-tile; W panel (K x 16, K <= 512) staged in LDS.
__global__ __launch_bounds__(256) void k_gemm_wmma(
    const float* __restrict__ A, int lda, const float* __restrict__ W, int ldw,
    const float* __restrict__ bias, float* __restrict__ C, int ldc,
    int M, int N, int K, int relu) {
  __shared__ float sW[512 * 16];          // 32 KB of the 320 KB/WGP pool
  int tilesN = (N + 15) >> 4;
  int n0 = (blockIdx.x % tilesN) * 16;
  int bm = blockIdx.x / tilesN;
  for (int idx = threadIdx.x; idx < K * 16; idx += 256) {
    int k = idx >> 4, c = idx & 15;
    int n = n0 + c;
    if (n < N) async_load_f32_to_lds(&W[(size_t)k * ldw + n], &sW[idx]);
    else       sW[idx] = 0.f;
  }
  wait_async_all();
  __syncthreads();

  int tm = bm * 8 + (threadIdx.x >> 5);
  if (tm * 16 >= M) return;
  int lane = threadIdx.x & 31;
  int half = lane >> 4, l16 = lane & 15;
  int mrow = tm * 16 + l16;
  int ncol = n0 + l16;
  bool mok = mrow < M;
  bool evenlda = (lda & 1) == 0;
  const float* aRow = A + (size_t)mrow * lda;
  v8f acc = {0.f, 0.f, 0.f, 0.f, 0.f, 0.f, 0.f, 0.f};
#pragma unroll 4
  for (int kb = 0; kb < K; kb += 4) {
    int kk = kb + 2 * half;
    v2f a;
    if (mok && evenlda && (kk + 1) < K) {
      a = *(const v2f*)(aRow + kk);       // kk even -> 8B aligned when lda even
    } else {
      a[0] = (mok && kk < K)       ? aRow[kk]     : 0.f;
      a[1] = (mok && (kk + 1) < K) ? aRow[kk + 1] : 0.f;
    }
    v2f b;
    b[0] = (kk < K)       ? sW[kk * 16 + l16]       : 0.f;
    b[1] = ((kk + 1) < K) ? sW[(kk + 1) * 16 + l16] : 0.f;
    acc = __builtin_amdgcn_wmma_f32_16x16x4_f32(false, a, false, b, (short)0, acc, false, false);
  }
  float bb = (ncol < N) ? bias[ncol] : 0.f;
#pragma unroll
  for (int rr = 0; rr < 8; ++rr) {
    int m = tm * 16 + rr + 8 * half;
    if (m < M && ncol < N) {
      float v = acc[rr] + bb;
      if (relu) v = fmaxf(v, 0.f);
      C[(size_t)m * ldc + ncol] = v;
    }
  }
}

// ---------------- N=1 projections ----------------
__global__ __launch_bounds__(256) void k_rowdot(const float* __restrict__ A, int lda, int K,
                                                const float* __restrict__ w,
                                                const float* __restrict__ b,
                                                float* __restrict__ out, int M) {
  int m = blockIdx.x * 256 + threadIdx.x;
  if (m >= M) return;
  float s = b[0];
  for (int k = 0; k < K; ++k) s += A[(size_t)m * lda + k] * w[k];
  out[m] = s;
}

__global__ __launch_bounds__(256) void k_conv_out1(const float* __restrict__ agg,
                                                   const float* __restrict__ cnt,
                                                   const float* __restrict__ hroot,
                                                   const float* __restrict__ Wn,
                                                   const float* __restrict__ Wr,
                                                   const float* __restrict__ b,
                                                   float* __restrict__ out, int M) {
  int m = blockIdx.x * 256 + threadIdx.x;
  if (m >= M) return;
  float rinv = 1.f / fmaxf(cnt[m], 1.f);
  float s = 0.f, r = 0.f;
  for (int k = 0; k < 64; ++k) {
    s += agg[(size_t)m * 64 + k] * Wn[k];
    r += hroot[(size_t)m * 64 + k] * Wr[k];
  }
  out[m] = s * rinv + r + b[0];
}

// ---------------- xin assembly: [xnode(64) | h3(1) | xl1(1) | xl2(1)] ----------------
__global__ __launch_bounds__(256) void k_buildxin(const float* __restrict__ xnode,
                                                  const float* __restrict__ h3,
                                                  const float* __restrict__ xl1,
                                                  const float* __restrict__ xl2,
                                                  float* __restrict__ xin, int M) {
  int i = blockIdx.x * 256 + threadIdx.x;
  if (i >= M * 67) return;
  int m = i / 67, n = i - m * 67;
  float v;
  if (n < 64)       v = xnode[(size_t)m * 64 + n];
  else if (n == 64) v = h3[m];
  else if (n == 65) v = xl1[m];
  else              v = xl2[m];
  xin[i] = v;
}

// ---------------- softmax over 67 columns (per row) ----------------
__global__ __launch_bounds__(256) void k_softmax67(float* __restrict__ p, int M) {
  int m = blockIdx.x * 256 + threadIdx.x;
  if (m >= M) return;
  float* row = p + (size_t)m * 67;
  float mx = row[0];
  for (int k = 1; k < 67; ++k) mx = fmaxf(mx, row[k]);
  float s = 0.f;
  for (int k = 0; k < 67; ++k) { float e = expf(row[k] - mx); row[k] = e; s += e; }
  float inv = 1.f / s;
  for (int k = 0; k < 67; ++k) row[k] *= inv;
}

// y = xin*prob + xin (prob in-place in tmp)
__global__ __launch_bounds__(256) void k_attmix(float* __restrict__ tmp,
                                                const float* __restrict__ xin, int n) {
  int i = blockIdx.x * 256 + threadIdx.x;
  if (i >= n) return;
  float x = xin[i];
  tmp[i] = x * tmp[i] + x;
}

// Per-column sum / sum-of-squares: one block per column, LDS tree reduce.
__global__ __launch_bounds__(256) void k_colstats(const float* __restrict__ src, int M, int N,
                                                  float* __restrict__ sum,
                                                  float* __restrict__ ssq) {
  int n = blockIdx.x;
  float s = 0.f, q = 0.f;
  for (int m = threadIdx.x; m < M; m += 256) {
    float v = src[(size_t)m * N + n];
    s += v; q += v * v;
  }
  __shared__ float sh[256], qh[256];
  sh[threadIdx.x] = s; qh[threadIdx.x] = q;
  __syncthreads();
  for (int off = 128; off > 0; off >>= 1) {
    if (threadIdx.x < off) {
      sh[threadIdx.x] += sh[threadIdx.x + off];
      qh[threadIdx.x] += qh[threadIdx.x + off];
    }
    __syncthreads();
  }
  if (threadIdx.x == 0) { sum[n] = sh[0]; ssq[n] = qh[0]; }
}

// dst = g*(src-mean)*rsqrt(var+eps)+beta (+skip). Biased batch stats.
__global__ __launch_bounds__(256) void k_bn(float* __restrict__ dst, const float* __restrict__ src,
                                            const float* __restrict__ skip,
                                            const float* __restrict__ sum,
                                            const float* __restrict__ ssq,
                                            const float* __restrict__ g,
                                            const float* __restrict__ beta, int M, int N) {
  int i = blockIdx.x * 256 + threadIdx.x;
  if (i >= M * N) return;
  int n = i % N;
  float mean = sum[n] / (float)M;
  float var = ssq[n] / (float)M - mean * mean;
  float v = g[n] * (src[i] - mean) * rsqrtf(var + EPSBN) + beta[n];
  if (skip) v += skip[i];
  dst[i] = v;
}

static inline int cdiv(long long a, long long b) { return (int)((a + b - 1) / b); }

extern "C" void kernel_launch(void* const* d_in, const int* in_sizes, int n_in,
                              void* d_out, int out_size, void* d_ws, size_t ws_size,
                              hipStream_t stream) {
  (void)n_in; (void)out_size; (void)ws_size;
  const int M1 = 100000, M2 = 50000, M3 = 25000;

  const float* x     = (const float*)d_in[0];
  const float* xnode = (const float*)d_in[1];
  const int* ei1 = (const int*)d_in[2]; int E1 = in_sizes[2] / 2;
  const int* ei2 = (const int*)d_in[3]; int E2 = in_sizes[3] / 2;
  const int* ei3 = (const int*)d_in[4]; int E3 = in_sizes[4] / 2;
  // params flattened in jax pytree order (dict keys sorted):
  // atts, auto, bns, conv, last, last2, links
  const float* attW[2]    = {(const float*)d_in[5], (const float*)d_in[9]};
  const float* attB[2]    = {(const float*)d_in[6], (const float*)d_in[10]};
  const float* attBeta[2] = {(const float*)d_in[7], (const float*)d_in[11]};
  const float* attG[2]    = {(const float*)d_in[8], (const float*)d_in[12]};
  const float* autoW[5]; const float* autoB[5];
  for (int i = 0; i < 5; ++i) {
    autoW[i] = (const float*)d_in[13 + 2 * i];
    autoB[i] = (const float*)d_in[14 + 2 * i];
  }
  const float* bnB[6]; const float* bnG[6];
  for (int i = 0; i < 6; ++i) {
    bnB[i] = (const float*)d_in[23 + 2 * i];
    bnG[i] = (const float*)d_in[24 + 2 * i];
  }
  const float* convWn[3] = {(const float*)d_in[35], (const float*)d_in[38], (const float*)d_in[41]};
  const float* convWr[3] = {(const float*)d_in[36], (const float*)d_in[39], (const float*)d_in[42]};
  const float* convB[3]  = {(const float*)d_in[37], (const float*)d_in[40], (const float*)d_in[43]};
  const float* lastW  = (const float*)d_in[44];
  const float* lastB  = (const float*)d_in[45];
  const float* last2W = (const float*)d_in[46];
  const float* last2B = (const float*)d_in[47];
  const float* linkW[2] = {(const float*)d_in[48], (const float*)d_in[50]};
  const float* linkB[2] = {(const float*)d_in[49], (const float*)d_in[51]};

  float* ws  = (float*)d_ws;
  float* AGG = ws + F_R0;   float* H1  = ws + F_R1;   float* H2  = ws + F_R2;
  float* ZD  = ws + F_ZD;   float* ZE  = ws + F_ZE;   float* XIN = ws + F_XIN;
  float* TMP = ws + F_TMP;  float* ZF  = ws + F_ZF;   float* H3  = ws + F_H3;
  float* XL1 = ws + F_XL1;  float* XL2 = ws + F_XL2;  float* CNT = ws + F_CNT;
  float* SUM = ws + F_SUM;  float* SSQ = ws + F_SSQ;

  // grid helpers: conv grid = 4 N-tiles * ceil(M/128); gemm grid = tilesN * ceil(M/128)
  auto convGrid = [&](int M) { return 4 * cdiv(M, 128); };
  auto gemmGrid = [&](int M, int N) { return ((N + 15) / 16) * cdiv(M, 128); };

  // ---- conv hop 1 (200k src -> 100k dst) ----
  k_zero<<<cdiv((long long)M1 * 64, 256), 256, 0, stream>>>(AGG, M1 * 64);
  k_zero<<<cdiv(M1, 256), 256, 0, stream>>>(CNT, M1);
  k_scatter<<<cdiv((long long)E1 * 64, 256), 256, 0, stream>>>(ei1, ei1 + E1, x, AGG, CNT, E1);
  k_conv_wmma<<<convGrid(M1), 256, 0, stream>>>(AGG, CNT, x, convWn[0], convWr[0], convB[0], H1, M1, 1);
  k_rowdot<<<cdiv(M3, 256), 256, 0, stream>>>(H1, 64, 64, linkW[0], linkB[0], XL1, M3);

  // ---- conv hop 2 (100k -> 50k) ----
  k_zero<<<cdiv((long long)M2 * 64, 256), 256, 0, stream>>>(AGG, M2 * 64);
  k_zero<<<cdiv(M2, 256), 256, 0, stream>>>(CNT, M2);
  k_scatter<<<cdiv((long long)E2 * 64, 256), 256, 0, stream>>>(ei2, ei2 + E2, H1, AGG, CNT, E2);
  k_conv_wmma<<<convGrid(M2), 256, 0, stream>>>(AGG, CNT, H1, convWn[1], convWr[1], convB[1], H2, M2, 1);
  k_rowdot<<<cdiv(M3, 256), 256, 0, stream>>>(H2, 64, 64, linkW[1], linkB[1], XL2, M3);

  // ---- conv hop 3 (50k -> 25k, out width 1) ----
  k_zero<<<cdiv((long long)M3 * 64, 256), 256, 0, stream>>>(AGG, M3 * 64);
  k_zero<<<cdiv(M3, 256), 256, 0, stream>>>(CNT, M3);
  k_scatter<<<cdiv((long long)E3 * 64, 256), 256, 0, stream>>>(ei3, ei3 + E3, H2, AGG, CNT, E3);
  k_conv_out1<<<cdiv(M3, 256), 256, 0, stream>>>(AGG, CNT, H2, convWn[2], convWr[2], convB[2], H3, M3);

  // ---- xin = [xnode | h3 | xlink1 | xlink2] : [25000, 67] ----
  k_buildxin<<<cdiv((long long)M3 * 67, 256), 256, 0, stream>>>(xnode, H3, XL1, XL2, XIN, M3);

  // ---- 2x attention blocks: softmax(xin@W+b); xin = BN(xin*prob + xin) ----
  for (int a = 0; a < 2; ++a) {
    k_gemm_wmma<<<gemmGrid(M3, 67), 256, 0, stream>>>(
        XIN, 67, attW[a], 67, attB[a], TMP, 67, M3, 67, 67, 0);
    k_softmax67<<<cdiv(M3, 256), 256, 0, stream>>>(TMP, M3);
    k_attmix<<<cdiv((long long)M3 * 67, 256), 256, 0, stream>>>(TMP, XIN, M3 * 67);
    k_colstats<<<67, 256, 0, stream>>>(TMP, M3, 67, SUM, SSQ);
    k_bn<<<cdiv((long long)M3 * 67, 256), 256, 0, stream>>>(
        XIN, TMP, nullptr, SUM, SSQ, attG[a], attBeta[a], M3, 67);
  }

  // ---- autoencoder (reuse conv regions: zA=AGG, zB=H1, zC=H2) ----
  float* zA = AGG; float* zB = H1; float* zC = H2;
  // enc0: 67 -> 512
  k_gemm_wmma<<<gemmGrid(M3, 512), 256, 0, stream>>>(
      XIN, 67, autoW[0], 512, autoB[0], zA, 512, M3, 512, 67, 1);
  k_colstats<<<512, 256, 0, stream>>>(zA, M3, 512, SUM, SSQ);
  k_bn<<<cdiv((long long)M3 * 512, 256), 256, 0, stream>>>(zA, zA, nullptr, SUM, SSQ, bnG[0], bnB[0], M3, 512);
  // enc1: 512 -> 256
  k_gemm_wmma<<<gemmGrid(M3, 256), 256, 0, stream>>>(
      zA, 512, autoW[1], 256, autoB[1], zB, 256, M3, 256, 512, 1);
  k_colstats<<<256, 256, 0, stream>>>(zB, M3, 256, SUM, SSQ);
  k_bn<<<cdiv((long long)M3 * 256, 256), 256, 0, stream>>>(zB, zB, nullptr, SUM, SSQ, bnG[1], bnB[1], M3, 256);
  // enc2: 256 -> 128
  k_gemm_wmma<<<gemmGrid(M3, 128), 256, 0, stream>>>(
      zB, 256, autoW[2], 128, autoB[2], zC, 128, M3, 128, 256, 1);
  k_colstats<<<128, 256, 0, stream>>>(zC, M3, 128, SUM, SSQ);
  k_bn<<<cdiv((long long)M3 * 128, 256), 256, 0, stream>>>(zC, zC, nullptr, SUM, SSQ, bnG[2], bnB[2], M3, 128);
  // dec0: 128 -> 256, skip += zB
  k_gemm_wmma<<<gemmGrid(M3, 256), 256, 0, stream>>>(
      zC, 128, autoW[3], 256, autoB[3], ZD, 256, M3, 256, 128, 1);
  k_colstats<<<256, 256, 0, stream>>>(ZD, M3, 256, SUM, SSQ);
  k_bn<<<cdiv((long long)M3 * 256, 256), 256, 0, stream>>>(ZD, ZD, zB, SUM, SSQ, bnG[3], bnB[3], M3, 256);
  // dec1: 256 -> 512, skip += zA
  k_gemm_wmma<<<gemmGrid(M3, 512), 256, 0, stream>>>(
      ZD, 256, autoW[4], 512, autoB[4], ZE, 512, M3, 512, 256, 1);
  k_colstats<<<512, 256, 0, stream>>>(ZE, M3, 512, SUM, SSQ);
  k_bn<<<cdiv((long long)M3 * 512, 256), 256, 0, stream>>>(ZE, ZE, zA, SUM, SSQ, bnG[4], bnB[4], M3, 512);
  // last2: 512 -> 67, relu fused, then BN + skip xin
  k_gemm_wmma<<<gemmGrid(M3, 67), 256, 0, stream>>>(
      ZE, 512, last2W, 67, last2B, ZF, 67, M3, 67, 512, 1);
  k_colstats<<<67, 256, 0, stream>>>(ZF, M3, 67, SUM, SSQ);
  k_bn<<<cdiv((long long)M3 * 67, 256), 256, 0, stream>>>(ZF, ZF, XIN, SUM, SSQ, bnG[5], bnB[5], M3, 67);
  // final head: 67 -> 1
  k_rowdot<<<cdiv(M3, 256), 256, 0, stream>>>(ZF, 67, 67, lastW, lastB, (float*)d_out, M3);
}